// MediumLSTM_54649163874401
// MI455X (gfx1250) — compile-verified
//
#include <hip/hip_runtime.h>
#include <math.h>

typedef float v2f __attribute__((ext_vector_type(2)));
typedef float v4f __attribute__((ext_vector_type(4)));
typedef float v8f __attribute__((ext_vector_type(8)));

#define WMMA_F32(A, B, C) \
  __builtin_amdgcn_wmma_f32_16x16x4_f32(false, (A), false, (B), (short)0, (C), false, false)

#define T_ 256   // timesteps
#define I_ 14    // input size
#define H_ 64    // hidden size
#define G_ 256   // 4*H (gate columns: i|f|g|o)
#define MT 16    // batch rows per workgroup
#define XS 20    // x tile row stride (dwords)
#define GS 20    // gates buffer column stride (col-major [256][20])
#define HS 68    // h buffer row stride (dwords) -> conflict-free b64 frag loads

// Branch-free fast transcendentals (hardware v_exp_f32 path, saturating at +-inf)
__device__ __forceinline__ float sigm(float v) {
  return __fdividef(1.0f, 1.0f + __expf(-v));
}
__device__ __forceinline__ float tanh_fast(float v) {
  return 1.0f - __fdividef(2.0f, 1.0f + __expf(2.0f * v));
}

__global__ __launch_bounds__(512)
void lstm2_fused(const float* __restrict__ x,
                 const float* __restrict__ wih0, const float* __restrict__ whh0,
                 const float* __restrict__ bih0, const float* __restrict__ bhh0,
                 const float* __restrict__ wih1, const float* __restrict__ whh1,
                 const float* __restrict__ bih1, const float* __restrict__ bhh1,
                 const float* __restrict__ wfc1, const float* __restrict__ bfc1,
                 const float* __restrict__ wfc2, const float* __restrict__ bfc2,
                 float* __restrict__ out)
{
  __shared__ float xb[MT * XS];   // x_t tile, K-padded (cols 14..19 stay zero)
  __shared__ float gb[G_ * GS];   // gate pre-activations, col-major [n][m]
  __shared__ float h1b[MT * HS];  // h1 state [m][u]
  __shared__ float h2b[MT * HS];  // h2 state [m][u]

  const int tid  = threadIdx.x;
  const int lane = tid & 31;
  const int wv   = tid >> 5;      // wave id == N-tile (16 gate columns)
  const int ln   = lane & 15;     // m (A frag) / n-within-tile (B,C frags)
  const int hf   = lane >> 4;     // K-half selector
  const int b0   = blockIdx.x * MT;
  const int ncol = wv * 16 + ln;  // global gate column 0..255

  // ---- resident weight B-fragments: b[j] = W[ncol][4*kk + 2*hf + j] ----
  v2f fhh0[16], fih1[16], fhh1[16], fih0[4];
#pragma unroll
  for (int kk = 0; kk < 16; ++kk) {
    const int k = 4 * kk + 2 * hf;
    fhh0[kk] = *(const v2f*)(whh0 + ncol * H_ + k);
    fih1[kk] = *(const v2f*)(wih1 + ncol * H_ + k);
    fhh1[kk] = *(const v2f*)(whh1 + ncol * H_ + k);
  }
#pragma unroll
  for (int kk = 0; kk < 4; ++kk) {
    const int k = 4 * kk + 2 * hf;
    v2f v; v[0] = 0.0f; v[1] = 0.0f;
    if (k < I_ - 1) v = *(const v2f*)(wih0 + ncol * I_ + k);  // k=14 pad -> 0
    fih0[kk] = v;
  }
  const float bias0 = bih0[ncol] + bhh0[ncol];
  const float bias1 = bih1[ncol] + bhh1[ncol];

  // elementwise-phase cell mapping: (em,eu) and (em+8,eu); c-state in regs
  const int em = tid >> 6;   // 0..7
  const int eu = tid & 63;   // hidden unit
  float c1a = 0.0f, c1b = 0.0f, c2a = 0.0f, c2b = 0.0f;

  for (int i = tid; i < MT * XS; i += 512) xb[i] = 0.0f;
  for (int i = tid; i < MT * HS; i += 512) { h1b[i] = 0.0f; h2b[i] = 0.0f; }

  // ---- pipelined x loader: stage x[0] now; xv holds x[t+1] across the loop ----
  const bool xload = (tid < MT * I_);
  const int  xm = tid / I_, xc = tid - (tid / I_) * I_;
  const float* xrow = x + (size_t)(b0 + xm) * (T_ * I_) + xc;
  if (xload) xb[xm * XS + xc] = xrow[0];
  float xv = xload ? xrow[I_] : 0.0f;   // x[1]

  __syncthreads();

  for (int t = 0; t < T_; ++t) {
    // issue x[t+2] load now; latency hides behind the phase-A WMMA chain
    float xv2 = 0.0f;
    if (xload && (t + 2 < T_)) xv2 = xrow[(t + 2) * I_];

    // ---- phase A: layer-1 gates = bias0 + x_t@Wih0^T + h1@Whh0^T ----
    {
      v2f ax[4], ah[16];
#pragma unroll
      for (int kk = 0; kk < 4; ++kk)
        ax[kk] = *(const v2f*)(xb + ln * XS + 4 * kk + 2 * hf);
#pragma unroll
      for (int kk = 0; kk < 16; ++kk)
        ah[kk] = *(const v2f*)(h1b + ln * HS + 4 * kk + 2 * hf);
      v8f acc = {bias0, bias0, bias0, bias0, bias0, bias0, bias0, bias0};
#pragma unroll
      for (int kk = 0; kk < 4; ++kk)  acc = WMMA_F32(ax[kk], fih0[kk], acc);
#pragma unroll
      for (int kk = 0; kk < 16; ++kk) acc = WMMA_F32(ah[kk], fhh0[kk], acc);
      float* gp = gb + ncol * GS + hf * 8;           // C frag: M = v + 8*hf
      *(v4f*)gp       = *(v4f*)&acc;
      *(v4f*)(gp + 4) = *((v4f*)&acc + 1);
    }
    __syncthreads();

    // ---- phase B: layer-1 elementwise -> c1 (regs), h1 -> LDS ----
    {
      float ig = gb[(eu      ) * GS + em], fg = gb[(eu +  64) * GS + em];
      float gg = gb[(eu + 128) * GS + em], og = gb[(eu + 192) * GS + em];
      c1a = sigm(fg) * c1a + sigm(ig) * tanh_fast(gg);
      h1b[em * HS + eu] = sigm(og) * tanh_fast(c1a);
      ig = gb[(eu      ) * GS + em + 8]; fg = gb[(eu +  64) * GS + em + 8];
      gg = gb[(eu + 128) * GS + em + 8]; og = gb[(eu + 192) * GS + em + 8];
      c1b = sigm(fg) * c1b + sigm(ig) * tanh_fast(gg);
      h1b[(em + 8) * HS + eu] = sigm(og) * tanh_fast(c1b);
    }
    __syncthreads();

    // ---- phase C: layer-2 gates = bias1 + h1@Wih1^T + h2@Whh1^T ----
    {
      v8f acc = {bias1, bias1, bias1, bias1, bias1, bias1, bias1, bias1};
      v2f a1[16];
#pragma unroll
      for (int kk = 0; kk < 16; ++kk)
        a1[kk] = *(const v2f*)(h1b + ln * HS + 4 * kk + 2 * hf);
#pragma unroll
      for (int kk = 0; kk < 16; ++kk) acc = WMMA_F32(a1[kk], fih1[kk], acc);
      v2f a2[16];
#pragma unroll
      for (int kk = 0; kk < 16; ++kk)
        a2[kk] = *(const v2f*)(h2b + ln * HS + 4 * kk + 2 * hf);
#pragma unroll
      for (int kk = 0; kk < 16; ++kk) acc = WMMA_F32(a2[kk], fhh1[kk], acc);
      float* gp = gb + ncol * GS + hf * 8;
      *(v4f*)gp       = *(v4f*)&acc;
      *(v4f*)(gp + 4) = *((v4f*)&acc + 1);
    }
    __syncthreads();

    // ---- phase D: layer-2 elementwise + stage x[t+1] tile ----
    if (xload && (t + 1 < T_)) xb[xm * XS + xc] = xv;
    {
      float ig = gb[(eu      ) * GS + em], fg = gb[(eu +  64) * GS + em];
      float gg = gb[(eu + 128) * GS + em], og = gb[(eu + 192) * GS + em];
      c2a = sigm(fg) * c2a + sigm(ig) * tanh_fast(gg);
      h2b[em * HS + eu] = sigm(og) * tanh_fast(c2a);
      ig = gb[(eu      ) * GS + em + 8]; fg = gb[(eu +  64) * GS + em + 8];
      gg = gb[(eu + 128) * GS + em + 8]; og = gb[(eu + 192) * GS + em + 8];
      c2b = sigm(fg) * c2b + sigm(ig) * tanh_fast(gg);
      h2b[(em + 8) * HS + eu] = sigm(og) * tanh_fast(c2b);
    }
    xv = xv2;
    __syncthreads();
  }

  // ---- FC head: z = relu(h2_T @ Wfc1^T + b1); out = z @ Wfc2^T + b2 ----
  {
    float a0 = bfc1[eu], a1 = bfc1[eu];
    for (int u = 0; u < H_; ++u) {
      const float w = wfc1[eu * H_ + u];
      a0 += h2b[em * HS + u] * w;
      a1 += h2b[(em + 8) * HS + u] * w;
    }
    h1b[em * HS + eu]       = fmaxf(a0, 0.0f);   // reuse h1 buffer for z
    h1b[(em + 8) * HS + eu] = fmaxf(a1, 0.0f);
  }
  __syncthreads();
  if (tid < MT * 2) {
    const int m = tid >> 1, c = tid & 1;
    float a = bfc2[c];
    for (int u = 0; u < H_; ++u) a += h1b[m * HS + u] * wfc2[c * H_ + u];
    out[(size_t)(b0 + m) * 2 + c] = a;
  }
}

extern "C" void kernel_launch(void* const* d_in, const int* in_sizes, int n_in,
                              void* d_out, int out_size, void* d_ws, size_t ws_size,
                              hipStream_t stream) {
  (void)n_in; (void)out_size; (void)d_ws; (void)ws_size;
  const float* xp   = (const float*)d_in[0];
  const float* wih0 = (const float*)d_in[1];
  const float* whh0 = (const float*)d_in[2];
  const float* bih0 = (const float*)d_in[3];
  const float* bhh0 = (const float*)d_in[4];
  const float* wih1 = (const float*)d_in[5];
  const float* whh1 = (const float*)d_in[6];
  const float* bih1 = (const float*)d_in[7];
  const float* bhh1 = (const float*)d_in[8];
  const float* wfc1 = (const float*)d_in[9];
  const float* bfc1 = (const float*)d_in[10];
  const float* wfc2 = (const float*)d_in[11];
  const float* bfc2 = (const float*)d_in[12];
  float* outp = (float*)d_out;

  const int B = in_sizes[0] / (T_ * I_);   // 8192
  const int grid = B / MT;                 // 512 workgroups of 16 batch rows

  lstm2_fused<<<grid, 512, 0, stream>>>(xp, wih0, whh0, bih0, bhh0,
                                        wih1, whh1, bih1, bhh1,
                                        wfc1, bfc1, wfc2, bfc2, outp);
}